// VectorQuantizer_31009663877330
// MI455X (gfx1250) — compile-verified
//
#include <hip/hip_runtime.h>

typedef float v2f __attribute__((ext_vector_type(2)));
typedef float v8f __attribute__((ext_vector_type(8)));

#define D_DIM 256
#define ZSTRIDE 260   // padded LDS row stride (floats): conflict-free, 16B aligned
#define BETA 0.25f

// One block = 16 tokens (one WMMA M-tile). 8 waves; wave w scans codes
// [w*nE/8, (w+1)*nE/8) processing TWO 16-wide N-tiles at once so each
// A load from LDS feeds two independent v_wmma_f32_16x16x4_f32 chains.
__global__ __launch_bounds__(256) void vq_argmin_kernel(
    const float* __restrict__ z, const float* __restrict__ emb,
    float* __restrict__ out, int Ntot, int nE) {
  __shared__ float zsh[16 * ZSTRIDE];
  __shared__ float waveD[8][16];
  __shared__ int   waveI[8][16];
  __shared__ int   finalIdx[16];

  const int tid  = threadIdx.x;
  const int lane = tid & 31;
  const int wv   = tid >> 5;
  const int rowG0 = blockIdx.x * 16;

  // ---- stage z tile (16 x 256 f32) into LDS with float4 copies ----
  {
    const float4* zg = reinterpret_cast<const float4*>(z + (size_t)rowG0 * D_DIM);
    #pragma unroll
    for (int j = 0; j < 4; ++j) {
      int i = tid + j * 256;      // 1024 float4s total
      int r = i >> 6;             // 64 float4 per row
      int c = i & 63;
      float4 v = zg[(size_t)r * 64 + c];
      *reinterpret_cast<float4*>(&zsh[r * ZSTRIDE + c * 4]) = v;
    }
  }
  __syncthreads();

  // WMMA fp32 16x16x4 lane mapping:
  //  A (16x4):  lane -> M = lane&15 ; VGPR pair holds K = {0,1} (lanes 0-15) / {2,3} (lanes 16-31)
  //  B (4x16):  lane -> N = lane&15 ; same K-half split
  //  C (16x16): lanes 0-15 = rows 0-7 in acc[0..7]; lanes 16-31 = rows 8-15
  const int mRow  = lane & 15;
  const int kHalf = (lane >> 4) << 1;          // 0 or 2
  const int codesPerWave = nE >> 3;            // 128
  const int codeBase0 = wv * codesPerWave;

  float bestD[8];
  int   bestI[8];
  #pragma unroll
  for (int s = 0; s < 8; ++s) { bestD[s] = 3.4e38f; bestI[s] = 0x7fffffff; }

  for (int t = 0; t < codesPerWave; t += 32) { // two 16-wide N-tiles per pass
    const int code0 = codeBase0 + t + mRow;
    const int code1 = code0 + 16;
    const float* erow0 = emb + (size_t)code0 * D_DIM;
    const float* erow1 = emb + (size_t)code1 * D_DIM;
    v8f acc0 = {};
    v8f acc1 = {};
    float en0 = 0.f;                           // partial ||e||^2 (this K-half)
    float en1 = 0.f;
    #pragma unroll 4
    for (int k = 0; k < D_DIM; k += 4) {
      v2f a  = *reinterpret_cast<const v2f*>(&zsh[mRow * ZSTRIDE + k + kHalf]);
      v2f b0 = *reinterpret_cast<const v2f*>(&erow0[k + kHalf]);
      v2f b1 = *reinterpret_cast<const v2f*>(&erow1[k + kHalf]);
      en0 += b0[0] * b0[0] + b0[1] * b0[1];
      en1 += b1[0] * b1[0] + b1[1] * b1[1];
      acc0 = __builtin_amdgcn_wmma_f32_16x16x4_f32(
          false, a, false, b0, (short)0, acc0, false, false);
      acc1 = __builtin_amdgcn_wmma_f32_16x16x4_f32(
          false, a, false, b1, (short)0, acc1, false, false);
    }
    en0 += __shfl_xor(en0, 16, 32);            // combine K-halves -> full ||e||^2
    en1 += __shfl_xor(en1, 16, 32);

    // dist' = ||e||^2 - 2*dot  (the ||z||^2 row constant cannot change argmin)
    #pragma unroll
    for (int s = 0; s < 8; ++s) {
      float d0 = en0 - 2.0f * acc0[s];
      if (d0 < bestD[s] || (d0 == bestD[s] && code0 < bestI[s])) {
        bestD[s] = d0; bestI[s] = code0;
      }
      float d1 = en1 - 2.0f * acc1[s];
      if (d1 < bestD[s] || (d1 == bestD[s] && code1 < bestI[s])) {
        bestD[s] = d1; bestI[s] = code1;
      }
    }
  }

  // ---- min+argmin across the 16 lanes of each half-wave, per row slot ----
  #pragma unroll
  for (int s = 0; s < 8; ++s) {
    float d = bestD[s]; int i = bestI[s];
    for (int mask = 8; mask >= 1; mask >>= 1) {
      float od = __shfl_xor(d, mask, 32);
      int   oi = __shfl_xor(i, mask, 32);
      if (od < d || (od == d && oi < i)) { d = od; i = oi; }
    }
    bestD[s] = d; bestI[s] = i;
  }
  if (lane == 0) {
    #pragma unroll
    for (int s = 0; s < 8; ++s) { waveD[wv][s] = bestD[s]; waveI[wv][s] = bestI[s]; }
  } else if (lane == 16) {
    #pragma unroll
    for (int s = 0; s < 8; ++s) { waveD[wv][8 + s] = bestD[s]; waveI[wv][8 + s] = bestI[s]; }
  }
  __syncthreads();

  const size_t lossOff = (size_t)Ntot * D_DIM;
  const size_t idxOff  = lossOff + (size_t)Ntot;

  // ---- cross-wave argmin + index output ----
  if (tid < 16) {
    float d = waveD[0][tid]; int i = waveI[0][tid];
    #pragma unroll
    for (int w = 1; w < 8; ++w) {
      float od = waveD[w][tid]; int oi = waveI[w][tid];
      if (od < d || (od == d && oi < i)) { d = od; i = oi; }
    }
    finalIdx[tid] = i;
    out[idxOff + rowG0 + tid] = (float)i;
  }
  __syncthreads();

  // ---- epilogue: z_q_st = z + (e - z), loss = (1+beta)*mean((e - z)^2) ----
  const int row = tid >> 4;                    // 16 threads per row
  const int c0  = tid & 15;
  const int idx = finalIdx[row];
  const float* e = emb + (size_t)idx * D_DIM;
  float* zq = out + (size_t)(rowG0 + row) * D_DIM;
  float part = 0.f;
  #pragma unroll
  for (int j = 0; j < 16; ++j) {
    int d = c0 + (j << 4);
    float ev = e[d];
    float zv = zsh[row * ZSTRIDE + d];
    float diff = ev - zv;
    part += diff * diff;
    zq[d] = zv + diff;                         // straight-through forward value
  }
  for (int mask = 8; mask >= 1; mask >>= 1)    // 16 consecutive lanes share a row
    part += __shfl_xor(part, mask, 32);
  if (c0 == 0)
    out[lossOff + rowG0 + row] = part * ((1.0f + BETA) / (float)D_DIM);
}

extern "C" void kernel_launch(void* const* d_in, const int* in_sizes, int n_in,
                              void* d_out, int out_size, void* d_ws, size_t ws_size,
                              hipStream_t stream) {
  const float* z   = (const float*)d_in[0];   // [64,2048,256] f32
  const float* emb = (const float*)d_in[1];   // [1024,256] f32
  float* out = (float*)d_out;                 // [zq | loss | idx] flat f32
  const int Ntot = in_sizes[0] / D_DIM;       // 131072 tokens
  const int nE   = in_sizes[1] / D_DIM;       // 1024 codes
  const int blocks = Ntot / 16;               // 8192
  vq_argmin_kernel<<<blocks, 256, 0, stream>>>(z, emb, out, Ntot, nE);
}